// PerceptulSchemas_85864986182031
// MI455X (gfx1250) — compile-verified
//
#include <hip/hip_runtime.h>
#include <hip/hip_bf16.h>

// dims
#define T_  16
#define B_  8
#define N_  8
#define H_  128
#define C_  32
#define HW_ 256
#define TAR_ 145
#define CTX_ 273
#define D_   896      // HEADS*H
#define CONV_FLAT_ 8192
#define MEM_IN_  8593
#define MEM_PAD_ 8596  // pad K to multiple of 4, pad region zeroed
#define G4H_ 512

typedef float v2f  __attribute__((ext_vector_type(2)));
typedef float v2fu __attribute__((ext_vector_type(2), aligned(4)));  // 4B-aligned load type
typedef float v8f  __attribute__((ext_vector_type(8)));

__device__ __forceinline__ float sigmoidf_(float x) { return 1.f / (1.f + expf(-x)); }

// shared_const[j] = Wo_b[j] + sum_d Wo_w[j,d]*Wv_b[d]
// (attention collapses: all 9 keys identical -> output = v = Wv_b, for any q)
__global__ void k_shared_const(const float* __restrict__ Wo_w, const float* __restrict__ Wo_b,
                               const float* __restrict__ Wv_b, float* __restrict__ sc) {
  int j = threadIdx.x;  // 128
  float acc = Wo_b[j];
  for (int d = 0; d < D_; ++d) acc += Wo_w[j * D_ + d] * Wv_b[d];
  sc[j] = acc;
}

// WhhT[n, kk, g] = Whh[n, g, kk]   (coalesced reads in the recurrence)
__global__ void k_whhT(const float* __restrict__ Whh, float* __restrict__ WhhT) {
  int i = blockIdx.x * blockDim.x + threadIdx.x;
  if (i >= N_ * G4H_ * H_) return;
  int n = i / (G4H_ * H_);
  int rem = i % (G4H_ * H_);
  int g = rem / H_;
  int kk = rem % H_;
  WhhT[((size_t)n * H_ + kk) * G4H_ + g] = Whh[i];
}

// gate[r, n, c] = sigmoid( sum_{x<145} tar[r,x]*td_w[n,c,x] + td_b[n,c] )
__global__ void k_gate(const float* __restrict__ tar, const float* __restrict__ td_w,
                       const float* __restrict__ td_b, float* __restrict__ gate) {
  int r = blockIdx.x;        // t*B + b
  int tid = threadIdx.x;     // n*32 + c, 256 threads
  const float* trow = tar + (size_t)r * TAR_;
  const float* w = td_w + (size_t)tid * CTX_;
  float acc = td_b[tid];
  for (int x = 0; x < TAR_; ++x) acc += trow[x] * w[x];
  gate[(size_t)r * (N_ * C_) + tid] = sigmoidf_(acc);
}

// mem[r,n, d*256+hw] = sum_c image[r,c,hw] * gate[r,n,c] * conv_w[d,c] + conv_b[d]
__global__ void k_conv(const float* __restrict__ image, const float* __restrict__ gate,
                       const float* __restrict__ conv_w, const float* __restrict__ conv_b,
                       float* __restrict__ mem) {
  int blk = blockIdx.x;       // r*N + n
  int r = blk / N_;
  int hw = threadIdx.x;       // 256
  __shared__ float s_gw[C_ * C_];   // gate[c] * conv_w[d,c]
  for (int i = threadIdx.x; i < C_ * C_; i += blockDim.x) {
    int d = i >> 5, c = i & 31;
    s_gw[i] = gate[(size_t)blk * C_ + c] * conv_w[d * C_ + c];
  }
  __syncthreads();
  float img[C_];
  const float* ib = image + (size_t)r * C_ * HW_ + hw;
  for (int c = 0; c < C_; ++c) img[c] = ib[c * HW_];
  float* out = mem + (size_t)blk * MEM_PAD_ + hw;
  for (int d = 0; d < C_; ++d) {
    float acc = conv_b[d];
    const float* gw = s_gw + d * C_;
    for (int c = 0; c < C_; ++c) acc += img[c] * gw[c];
    out[d * HW_] = acc;
  }
}

// mem tail per row: [8192..8319]=shared_const, [8320..8464]=tar, [8465..8595]=0 (incl. K-pad)
__global__ void k_tail(const float* __restrict__ tar, const float* __restrict__ sc,
                       float* __restrict__ mem) {
  int blk = blockIdx.x;  // r*N + n
  int r = blk / N_;
  int idx = threadIdx.x; // 512
  if (idx >= (MEM_PAD_ - CONV_FLAT_)) return;   // 404 entries
  float v;
  if (idx < H_)             v = sc[idx];
  else if (idx < H_ + TAR_) v = tar[(size_t)r * TAR_ + (idx - H_)];
  else                      v = 0.f;
  mem[(size_t)blk * MEM_PAD_ + CONV_FLAT_ + idx] = v;
}

// xg[r,n,g] = mem[r,n,:] . Wih[n,g,:] + bih[n,g] + bhh[n,g]
// One wave per 16M x 64N tile: 4 independent accumulator chains (ILP across the
// WMMA latency), A fragment reused x4 (1 A + 4 B loads per 4 WMMAs).
__global__ __launch_bounds__(32) void k_gemm(const float* __restrict__ mem,
                                             const float* __restrict__ Wih,
                                             const float* __restrict__ bih,
                                             const float* __restrict__ bhh,
                                             float* __restrict__ xg) {
  int tile = blockIdx.x;            // 512 tiles
  int gT = tile & 7;                // 8 tiles of 64 columns (512/64)
  int mT = (tile >> 3) & 7;         // 8 m-tiles (128/16)
  int n  = tile >> 6;               // 8 schemas
  int lane = threadIdx.x & 31;
  int lm = lane & 15;
  int hi = lane >> 4;               // 0: K+0,K+1  1: K+2,K+3  (ISA 16x4 A layout)

  const float* A = mem + ((size_t)(mT * 16 + lm) * N_ + n) * MEM_PAD_ + 2 * hi;   // 8B aligned
  const float* B0 = Wih + ((size_t)n * G4H_ + (gT * 64 +  0 + lm)) * MEM_IN_ + 2 * hi;
  const float* B1 = Wih + ((size_t)n * G4H_ + (gT * 64 + 16 + lm)) * MEM_IN_ + 2 * hi;
  const float* B2 = Wih + ((size_t)n * G4H_ + (gT * 64 + 32 + lm)) * MEM_IN_ + 2 * hi;
  const float* B3 = Wih + ((size_t)n * G4H_ + (gT * 64 + 48 + lm)) * MEM_IN_ + 2 * hi;

  v8f acc0 = {0.f,0.f,0.f,0.f,0.f,0.f,0.f,0.f};
  v8f acc1 = acc0, acc2 = acc0, acc3 = acc0;
  for (int k = 0; k < 8592; k += 4) {
    v2f a  = *(const v2f*)(A + k);
    v2f b0 = (v2f)(*(const v2fu*)(B0 + k));   // Wih rows are odd-stride: only 4B aligned
    v2f b1 = (v2f)(*(const v2fu*)(B1 + k));
    v2f b2 = (v2f)(*(const v2fu*)(B2 + k));
    v2f b3 = (v2f)(*(const v2fu*)(B3 + k));
    acc0 = __builtin_amdgcn_wmma_f32_16x16x4_f32(false, a, false, b0, (short)0, acc0, false, false);
    acc1 = __builtin_amdgcn_wmma_f32_16x16x4_f32(false, a, false, b1, (short)0, acc1, false, false);
    acc2 = __builtin_amdgcn_wmma_f32_16x16x4_f32(false, a, false, b2, (short)0, acc2, false, false);
    acc3 = __builtin_amdgcn_wmma_f32_16x16x4_f32(false, a, false, b3, (short)0, acc3, false, false);
  }
  // K tail: k=8592 only. mem pad (8593..8595) is zero, so only guard B.
  {
    v2f a = *(const v2f*)(A + 8592);               // in-bounds (MEM_PAD_), zeros past 8592
    float l0 = (B0 - 2 * hi)[8592];                // row element 8592 valid for all lanes
    float l1 = (B1 - 2 * hi)[8592];
    float l2 = (B2 - 2 * hi)[8592];
    float l3 = (B3 - 2 * hi)[8592];
    v2f b0; b0.x = hi ? 0.f : l0; b0.y = 0.f;      // no divergence, no OOB loads
    v2f b1; b1.x = hi ? 0.f : l1; b1.y = 0.f;
    v2f b2; b2.x = hi ? 0.f : l2; b2.y = 0.f;
    v2f b3; b3.x = hi ? 0.f : l3; b3.y = 0.f;
    acc0 = __builtin_amdgcn_wmma_f32_16x16x4_f32(false, a, false, b0, (short)0, acc0, false, false);
    acc1 = __builtin_amdgcn_wmma_f32_16x16x4_f32(false, a, false, b1, (short)0, acc1, false, false);
    acc2 = __builtin_amdgcn_wmma_f32_16x16x4_f32(false, a, false, b2, (short)0, acc2, false, false);
    acc3 = __builtin_amdgcn_wmma_f32_16x16x4_f32(false, a, false, b3, (short)0, acc3, false, false);
  }

  v8f accs[4] = {acc0, acc1, acc2, acc3};
  for (int j = 0; j < 4; ++j) {
    int col = gT * 64 + 16 * j + lm;
    float bias = bih[n * G4H_ + col] + bhh[n * G4H_ + col];
    for (int v = 0; v < 8; ++v) {
      int row = mT * 16 + v + 8 * hi;              // ISA 16x16 f32 D layout
      xg[((size_t)row * N_ + n) * G4H_ + col] = accs[j][v] + bias;
    }
  }
}

// 16-step LSTM per (b,n); h/g in LDS, coalesced WhhT reads.
__global__ __launch_bounds__(512) void k_lstm(const float* __restrict__ xg,
                                              const float* __restrict__ WhhT,
                                              float* __restrict__ out) {
  int blk = blockIdx.x;   // b*N + n
  int n = blk % N_;
  int b = blk / N_;
  int tid = threadIdx.x;  // 512
  __shared__ float sh_h[H_];
  __shared__ float sh_g[G4H_];
  float c_reg = 0.f;
  if (tid < H_) sh_h[tid] = 0.f;
  __syncthreads();
  const float* Wt = WhhT + (size_t)n * H_ * G4H_;
  for (int t = 0; t < T_; ++t) {
    float acc = xg[(((size_t)t * B_ + b) * N_ + n) * G4H_ + tid];
    for (int kk = 0; kk < H_; ++kk) acc += sh_h[kk] * Wt[kk * G4H_ + tid];
    sh_g[tid] = acc;
    __syncthreads();
    if (tid < H_) {
      float ig = sigmoidf_(sh_g[tid]);
      float fg = sigmoidf_(sh_g[H_ + tid]);
      float gg = tanhf(sh_g[2 * H_ + tid]);
      float og = sigmoidf_(sh_g[3 * H_ + tid]);
      c_reg = fg * c_reg + ig * gg;
      float h = og * tanhf(c_reg);
      sh_h[tid] = h;
      out[((size_t)t * B_ + b) * (N_ * H_) + n * H_ + tid] = h;   // outs (T,B,N*H)
    }
    __syncthreads();
  }
  if (tid < H_) {
    float* hT = out + (size_t)T_ * B_ * N_ * H_;
    float* cT = hT + (size_t)B_ * N_ * H_;
    hT[((size_t)b * N_ + n) * H_ + tid] = sh_h[tid];
    cT[((size_t)b * N_ + n) * H_ + tid] = c_reg;
  }
}

extern "C" void kernel_launch(void* const* d_in, const int* in_sizes, int n_in,
                              void* d_out, int out_size, void* d_ws, size_t ws_size,
                              hipStream_t stream) {
  const float* image  = (const float*)d_in[0];
  const float* tar    = (const float*)d_in[1];
  // d_in[2..6]: Wq_w, Wq_b, Wk_w, Wk_b, Wv_w — algebraically dead (prev_h == 0)
  const float* Wv_b   = (const float*)d_in[7];
  const float* Wo_w   = (const float*)d_in[8];
  const float* Wo_b   = (const float*)d_in[9];
  const float* td_w   = (const float*)d_in[10];
  const float* td_b   = (const float*)d_in[11];
  const float* conv_w = (const float*)d_in[12];
  const float* conv_b = (const float*)d_in[13];
  const float* Wih    = (const float*)d_in[14];
  const float* Whh    = (const float*)d_in[15];
  const float* bih    = (const float*)d_in[16];
  const float* bhh    = (const float*)d_in[17];
  float* out = (float*)d_out;

  char* ws = (char*)d_ws;
  size_t off = 0;
  float* mem  = (float*)(ws + off); off += (size_t)T_ * B_ * N_ * MEM_PAD_ * sizeof(float); // 35.2MB
  float* gate = (float*)(ws + off); off += (size_t)T_ * B_ * N_ * C_ * sizeof(float);
  float* xg   = (float*)(ws + off); off += (size_t)T_ * B_ * N_ * G4H_ * sizeof(float);
  float* WhhT = (float*)(ws + off); off += (size_t)N_ * H_ * G4H_ * sizeof(float);
  float* sc   = (float*)(ws + off); off += 256 * sizeof(float);

  hipLaunchKernelGGL(k_shared_const, dim3(1), dim3(H_), 0, stream, Wo_w, Wo_b, Wv_b, sc);
  hipLaunchKernelGGL(k_whhT, dim3((N_ * G4H_ * H_ + 255) / 256), dim3(256), 0, stream, Whh, WhhT);
  hipLaunchKernelGGL(k_gate, dim3(T_ * B_), dim3(N_ * C_), 0, stream, tar, td_w, td_b, gate);
  hipLaunchKernelGGL(k_conv, dim3(T_ * B_ * N_), dim3(HW_), 0, stream, image, gate, conv_w, conv_b, mem);
  hipLaunchKernelGGL(k_tail, dim3(T_ * B_ * N_), dim3(512), 0, stream, tar, sc, mem);
  hipLaunchKernelGGL(k_gemm, dim3(512), dim3(32), 0, stream, mem, Wih, bih, bhh, xg);
  hipLaunchKernelGGL(k_lstm, dim3(B_ * N_), dim3(512), 0, stream, xg, WhhT, out);
}